// YOLOScript_46643344835185
// MI455X (gfx1250) — compile-verified
//
#include <hip/hip_runtime.h>

// ---------------- problem constants ----------------
#define BATCH   32
#define NBOX    10647          // 3*(13*13 + 26*26 + 52*52)
#define N0      507            // level0 boxes
#define N01     2535           // level0+level1 boxes
#define NCLS    80
#define TOPK    300
#define TPAD    304            // TOPK padded to 16
#define NTILE   19             // TPAD/16
#define BS      512            // nms block threads (16 waves)

typedef float v2f __attribute__((ext_vector_type(2)));
typedef float v8f __attribute__((ext_vector_type(8)));

// anchors gathered per level: lvl0 mask {6,7,8}, lvl1 {3,4,5}, lvl2 {0,1,2}
__constant__ float c_aw[9] = {116.f,156.f,373.f,  30.f,62.f, 59.f,  10.f,16.f,33.f};
__constant__ float c_ah[9] = { 90.f,198.f,326.f,  61.f,45.f,119.f,  13.f,30.f,23.f};

__device__ __forceinline__ float sigmf(float x) { return 1.0f / (1.0f + expf(-x)); }

// flat candidate index -> (level, anchor, spatial index, H*W, W)
__device__ __forceinline__ void locate(int n, int& lvl, int& a, int& p, int& HW, int& Wd)
{
    int rem;
    if (n < N0)       { lvl = 0; rem = n;       Wd = 13; }
    else if (n < N01) { lvl = 1; rem = n - N0;  Wd = 26; }
    else              { lvl = 2; rem = n - N01; Wd = 52; }
    HW = Wd * Wd;
    a  = rem / HW;
    p  = rem - a * HW;
}

// ---------------------------------------------------------------------------
// Kernel 1: streaming score pass.  score = sig(obj)*sig(max cls logit),
// thresholded at 0.5; cls = first argmax.  One thread per (b, n).
// Bandwidth-bound: ~110 MB read -> ~5us at 23.3 TB/s.
// ---------------------------------------------------------------------------
__global__ void yolo_score_kernel(const float* __restrict__ f0,
                                  const float* __restrict__ f1,
                                  const float* __restrict__ f2,
                                  float* __restrict__ score,
                                  int*   __restrict__ cls)
{
    int gid = blockIdx.x * blockDim.x + threadIdx.x;
    if (gid >= BATCH * NBOX) return;
    int b = gid / NBOX;
    int n = gid - b * NBOX;

    int lvl, a, p, HW, Wd;
    locate(n, lvl, a, p, HW, Wd);
    const float* base = (lvl == 0) ? f0 : (lvl == 1) ? f1 : f2;
    const float* f = base + (size_t)(b * 255 + a * 85) * HW + p;

    float obj = f[4 * HW];
    __builtin_prefetch(f + 5 * HW, 0, 1);   // global_prefetch_b8 ahead of cls scan

    float mv = f[5 * HW];
    int   mi = 0;
    for (int e = 1; e < NCLS; ++e) {
        float v = f[(5 + e) * HW];
        if (v > mv) { mv = v; mi = e; }     // strict > keeps first max (argmax semantics)
    }
    float s = sigmf(obj) * sigmf(mv);       // max of sigmoids == sigmoid of max
    if (s < 0.5f) s = 0.0f;                 // CONF_THRES
    score[gid] = s;
    cls[gid]   = mi;
}

// ---------------------------------------------------------------------------
// Kernel 2: per-image top-k + decode + IoU(WMMA area outer-sum) + greedy NMS.
// One block (512 thr = 16 waves) per image; working set stays in LDS.
// ---------------------------------------------------------------------------
struct P1 {                     // phase 1: top-k selection
    float sc[NBOX + 9];         // padded scores
    float rv[BS];               // reduction values
    int   ri[BS];               // reduction indices
};
struct P2 {                     // phase 2: boxes + suppression bitmask
    float    bx[TPAD][4];       // xyxy * 416 (no class offset)
    float    area[TPAD];
    unsigned S[TOPK * 10];      // S[i][w]: bit j set if iou(i,j) > 0.3
    unsigned keep[10];
};
union SMem { P1 p1; P2 p2; };

__global__ __launch_bounds__(BS)
void yolo_nms_kernel(const float* __restrict__ f0,
                     const float* __restrict__ f1,
                     const float* __restrict__ f2,
                     const float* __restrict__ score,
                     const int*   __restrict__ cls,
                     float*       __restrict__ out)
{
    __shared__ SMem u;
    __shared__ float tsc[TPAD];   // selected scores (descending)
    __shared__ int   tix[TPAD];   // selected flat indices
    __shared__ int   tcl[TPAD];   // selected class ids

    const int tid = threadIdx.x;
    const int b   = blockIdx.x;

    // ---- phase 1: load scores, iterative top-300 argmax (ties -> lower idx) ----
    for (int n = tid; n < NBOX; n += BS)
        u.p1.sc[n] = score[(size_t)b * NBOX + n];
    __syncthreads();

    for (int k = 0; k < TOPK; ++k) {
        float bv = -1.0f; int bi = NBOX;
        for (int n = tid; n < NBOX; n += BS) {      // ascending -> first max kept
            float v = u.p1.sc[n];
            if (v > bv) { bv = v; bi = n; }
        }
        u.p1.rv[tid] = bv; u.p1.ri[tid] = bi;
        __syncthreads();
        for (int s = BS / 2; s > 0; s >>= 1) {
            if (tid < s) {
                float v2 = u.p1.rv[tid + s]; int i2 = u.p1.ri[tid + s];
                if (v2 > u.p1.rv[tid] || (v2 == u.p1.rv[tid] && i2 < u.p1.ri[tid])) {
                    u.p1.rv[tid] = v2; u.p1.ri[tid] = i2;
                }
            }
            __syncthreads();
        }
        if (tid == 0) {
            int w = u.p1.ri[0];
            tix[k] = w; tsc[k] = u.p1.rv[0];
            u.p1.sc[w] = -2.0f;                     // remove from candidates
        }
        __syncthreads();
    }

    // ---- phase 2: decode the 300 selected boxes (re-read raw features) ----
    if (tid < TPAD) {
        if (tid < TOPK) {
            int n = tix[tid];
            int lvl, a, p, HW, Wd;
            locate(n, lvl, a, p, HW, Wd);
            const float* base = (lvl == 0) ? f0 : (lvl == 1) ? f1 : f2;
            const float* f = base + (size_t)(b * 255 + a * 85) * HW + p;
            int gy = p / Wd, gx = p - gy * Wd;
            float invG = 1.0f / (float)Wd;
            float x = (sigmf(f[0])      + (float)gx) * invG;
            float y = (sigmf(f[HW])     + (float)gy) * invG;
            float w = expf(f[2 * HW]) * c_aw[lvl * 3 + a] * (1.0f / 416.0f);
            float h = expf(f[3 * HW]) * c_ah[lvl * 3 + a] * (1.0f / 416.0f);
            float x1 = (x - 0.5f * w) * 416.0f, y1 = (y - 0.5f * h) * 416.0f;
            float x2 = (x + 0.5f * w) * 416.0f, y2 = (y + 0.5f * h) * 416.0f;
            u.p2.bx[tid][0] = x1; u.p2.bx[tid][1] = y1;
            u.p2.bx[tid][2] = x2; u.p2.bx[tid][3] = y2;
            u.p2.area[tid]  = (x2 - x1) * (y2 - y1);
            tcl[tid] = cls[(size_t)b * NBOX + n];
        } else {                                    // pad rows 300..303
            u.p2.bx[tid][0] = 0.f; u.p2.bx[tid][1] = 0.f;
            u.p2.bx[tid][2] = 0.f; u.p2.bx[tid][3] = 0.f;
            u.p2.area[tid]  = 0.f;
            tcl[tid] = 0; tsc[tid] = 0.f;
        }
    }
    __syncthreads();

    for (int i = tid; i < TOPK * 10; i += BS) u.p2.S[i] = 0u;
    __syncthreads();

    // ---- suppression matrix: WMMA rank-2 outer-sum for area_i + area_j,
    //      VALU intersection, ds-atomicOr bit packing.  One wave per 16x16 tile.
    {
        const int lane = tid & 31;
        const int wv   = tid >> 5;
        for (int t = wv; t < NTILE * NTILE; t += (BS / 32)) {
            int ti = t / NTILE, tj = t - ti * NTILE;
            // A (16x4 f32): col K=0 = area_i, K=1 = 1, K=2..3 = 0
            // B (4x16 f32): row K=0 = 1,     K=1 = area_j, K=2..3 = 0
            v2f A, Bv;
            if (lane < 16) {
                A.x  = u.p2.area[ti * 16 + lane]; A.y  = 1.0f;      // K=0,1
                Bv.x = 1.0f; Bv.y = u.p2.area[tj * 16 + lane];      // K=0,1
            } else {
                A.x = 0.0f; A.y = 0.0f; Bv.x = 0.0f; Bv.y = 0.0f;   // K=2,3
            }
            v8f Cz = {0.f,0.f,0.f,0.f,0.f,0.f,0.f,0.f};
            v8f D = __builtin_amdgcn_wmma_f32_16x16x4_f32(
                        false, A, false, Bv, (short)0, Cz, false, false);
            // D layout: lanes 0-15 -> N=lane, M=v ; lanes 16-31 -> N=lane-16, M=v+8
            int nj = tj * 16 + (lane & 15);
            float offj = 832.0f * (float)tcl[nj];   // class offset = cls * 2 * 416
            float jx1 = u.p2.bx[nj][0] + offj, jy1 = u.p2.bx[nj][1] + offj;
            float jx2 = u.p2.bx[nj][2] + offj, jy2 = u.p2.bx[nj][3] + offj;
            int mbase = ti * 16 + ((lane < 16) ? 0 : 8);
#pragma unroll
            for (int v = 0; v < 8; ++v) {
                int mi = mbase + v;
                float offi = 832.0f * (float)tcl[mi];
                float lx = fmaxf(u.p2.bx[mi][0] + offi, jx1);
                float ly = fmaxf(u.p2.bx[mi][1] + offi, jy1);
                float rx = fminf(u.p2.bx[mi][2] + offi, jx2);
                float ry = fminf(u.p2.bx[mi][3] + offi, jy2);
                float inter = fmaxf(rx - lx, 0.0f) * fmaxf(ry - ly, 0.0f);
                float iou   = inter / (D[v] - inter + 1e-6f);   // D[v] = area_i+area_j
                if (iou > 0.3f && mi < TOPK && nj < TOPK)
                    atomicOr(&u.p2.S[mi * 10 + (nj >> 5)], 1u << (nj & 31));
            }
        }
    }
    __syncthreads();

    // ---- greedy sequential NMS over the precomputed bitmask ----
    if (tid == 0) {
        unsigned keep[10];
        for (int w = 0; w < 10; ++w) keep[w] = 0xFFFFFFFFu;
        for (int i = 0; i < TOPK; ++i) {
            if ((keep[i >> 5] >> (i & 31)) & 1u) {
                int iw = i >> 5, ib = i & 31;
                for (int w = iw; w < 10; ++w) {
                    unsigned s = u.p2.S[i * 10 + w];
                    if (w == iw) s &= (ib == 31) ? 0u : (0xFFFFFFFFu << (ib + 1)); // only j > i
                    keep[w] &= ~s;
                }
            }
        }
        for (int w = 0; w < 10; ++w) u.p2.keep[w] = keep[w];
    }
    __syncthreads();

    // ---- emit [TOPK, 6] = (x1,y1,x2,y2,score,cls) * valid ----
    for (int k = tid; k < TOPK; k += BS) {
        bool  kb = (u.p2.keep[k >> 5] >> (k & 31)) & 1u;
        float sc = tsc[k];
        float m  = (kb && sc > 0.0f) ? 1.0f : 0.0f;
        float* o = out + ((size_t)b * TOPK + k) * 6;
        o[0] = u.p2.bx[k][0] * m;
        o[1] = u.p2.bx[k][1] * m;
        o[2] = u.p2.bx[k][2] * m;
        o[3] = u.p2.bx[k][3] * m;
        o[4] = sc * m;
        o[5] = (float)tcl[k] * m;
    }
}

// ---------------------------------------------------------------------------
extern "C" void kernel_launch(void* const* d_in, const int* in_sizes, int n_in,
                              void* d_out, int out_size, void* d_ws, size_t ws_size,
                              hipStream_t stream)
{
    (void)in_sizes; (void)n_in; (void)out_size; (void)ws_size;
    const float* f0 = (const float*)d_in[0];   // [32,255,13,13]
    const float* f1 = (const float*)d_in[1];   // [32,255,26,26]
    const float* f2 = (const float*)d_in[2];   // [32,255,52,52]

    float* score = (float*)d_ws;                                        // [32,10647] f32
    int*   clsid = (int*)((char*)d_ws + (size_t)BATCH * NBOX * 4);      // [32,10647] i32
    float* out   = (float*)d_out;                                       // [32,300,6] f32

    int total = BATCH * NBOX;
    yolo_score_kernel<<<(total + 255) / 256, 256, 0, stream>>>(f0, f1, f2, score, clsid);
    yolo_nms_kernel<<<BATCH, BS, 0, stream>>>(f0, f1, f2, score, clsid, out);
}